// UCBNorm_66735201845900
// MI455X (gfx1250) — compile-verified
//
#include <hip/hip_runtime.h>
#include <cstdint>

#define EPSF        0.001f
#define KMIX        8
#define DDIM        1024
#define BLROWS      8192        // B*L = 4*2048
#define TPB         256         // 8 wave32s
#define COLS        256         // columns per block tile
#define RPS         4           // rows per pipeline step (4*256 floats = 4KB)
#define GRID_X      (DDIM / COLS)       // 4
#define GRID_Y      64
#define ROWS_PB     (BLROWS / GRID_Y)   // 128 rows per block
#define NSTEPS      (ROWS_PB / RPS)     // 32 pipeline steps

// Issue an async global->LDS copy of 16 bytes per lane (GVS mode:
// saddr = 64-bit base SGPR pair, vaddr = 32-bit byte offset VGPR,
// vdst = LDS byte address VGPR). Tracked by ASYNCcnt.
#define ASYNC_B128(ldsaddr, goff, base64)                                   \
    asm volatile("global_load_async_to_lds_b128 %0, %1, %2"                 \
                 :: "v"(ldsaddr), "v"(goff), "s"(base64) : "memory")

__global__ __launch_bounds__(TPB) void ucbnorm_kernel(
    const float* __restrict__ x,
    const float* __restrict__ mean,
    const float* __restrict__ variance,
    const float* __restrict__ prior,
    float* __restrict__ out)
{
    __shared__ float buf[2][RPS * COLS];   // 2 x 4KB double buffer

    const int tid  = threadIdx.x;
    const int c0   = blockIdx.x * COLS;
    const int d    = c0 + tid;             // this thread's fixed column
    const int row0 = blockIdx.y * ROWS_PB;

    // ---- hoisted k-only constants: softmax(prior) over K (uniform) ----
    float pr[KMIX];
    {
        float m = -1e30f;
        #pragma unroll
        for (int k = 0; k < KMIX; ++k) { pr[k] = prior[k]; m = fmaxf(m, pr[k]); }
        float s = 0.0f;
        #pragma unroll
        for (int k = 0; k < KMIX; ++k) { pr[k] = exp2f((pr[k] - m) * 1.44269504f); s += pr[k]; }
        float inv = __builtin_amdgcn_rcpf(s);
        #pragma unroll
        for (int k = 0; k < KMIX; ++k) pr[k] *= inv;
    }

    // ---- hoisted (k,d) constants: 24 registers per thread ----
    // c1 = -0.5*log2(e)/(softplus(var)+eps)   (exponent folded for v_exp_f32)
    // c2 = rsqrt(pr+eps)*rsqrt(softplus(var)+eps)
    float mk[KMIX], c1[KMIX], c2[KMIX];
    #pragma unroll
    for (int k = 0; k < KMIX; ++k) {
        mk[k]    = mean[k * DDIM + d];
        float v  = variance[k * DDIM + d];
        float sp = __logf(1.0f + __expf(v));   // softplus
        float ve = sp + EPSF;
        c1[k]    = -0.72134752f / ve;          // -0.5 * log2(e) / ve
        c2[k]    = rsqrtf(pr[k] + EPSF) * rsqrtf(ve);
    }

    // ---- async pipeline addressing ----
    const uint32_t row_in_step = (uint32_t)tid >> 6;        // 0..3
    const uint32_t col4        = ((uint32_t)tid & 63u) << 2; // float4 column
    const uint64_t xbase       = (uint64_t)(uintptr_t)x;

    // global byte offset of this lane's 16B chunk for step 0
    uint32_t g = (uint32_t)(((uint32_t)(row0 + (int)row_in_step) * DDIM
                             + (uint32_t)c0 + col4) * 4u);
    const uint32_t gstep = (uint32_t)(RPS * DDIM * 4);      // 16 KiB per step

    const uint32_t ldsoff = (row_in_step * COLS + col4) * 4u;
    const uint32_t lds0 = (uint32_t)(uintptr_t)(&buf[0][0]) + ldsoff;
    const uint32_t lds1 = (uint32_t)(uintptr_t)(&buf[1][0]) + ldsoff;

    // prologue: fill buffer 0
    ASYNC_B128(lds0, g, xbase);

    for (int s = 0; s < NSTEPS; ++s) {
        if (s + 1 < NSTEPS) {
            const uint32_t nl = ((s + 1) & 1) ? lds1 : lds0;
            ASYNC_B128(nl, g + gstep, xbase);
            asm volatile("s_wait_asynccnt 0x1" ::: "memory"); // step s done (in-order)
        } else {
            asm volatile("s_wait_asynccnt 0x0" ::: "memory");
        }
        __syncthreads();   // all lanes' portions of buf[s&1] are visible

        const float* src = &buf[s & 1][0];
        #pragma unroll
        for (int r = 0; r < RPS; ++r) {
            const float xv = src[r * COLS + tid];   // ds_load_b32, conflict-free

            float pb[KMIX], db[KMIX];
            float denom = 0.0f;
            #pragma unroll
            for (int k = 0; k < KMIX; ++k) {
                const float diff = xv - mk[k];
                const float p    = pr[k] * exp2f(c1[k] * diff * diff); // v_exp_f32
                denom += p;
                pb[k] = p;
                db[k] = diff;
            }
            const float invden = __builtin_amdgcn_rcpf(denom + EPSF);

            float acc = 0.0f;
            #pragma unroll
            for (int k = 0; k < KMIX; ++k)
                acc = fmaf(pb[k] * db[k], c2[k], acc);
            acc *= invden;   // tau_k = p_k * invden folded out of the loop

            out[(size_t)(row0 + s * RPS + r) * DDIM + (size_t)d] = acc;
        }
        __syncthreads();   // buf[s&1] free for the load issued in iter s+1
        g += gstep;
    }
}

extern "C" void kernel_launch(void* const* d_in, const int* in_sizes, int n_in,
                              void* d_out, int out_size, void* d_ws, size_t ws_size,
                              hipStream_t stream) {
    const float* x        = (const float*)d_in[0];
    const float* mean     = (const float*)d_in[1];
    const float* variance = (const float*)d_in[2];
    const float* prior    = (const float*)d_in[3];
    float* out            = (float*)d_out;

    dim3 grid(GRID_X, GRID_Y, 1);
    dim3 block(TPB, 1, 1);
    ucbnorm_kernel<<<grid, block, 0, stream>>>(x, mean, variance, prior, out);
}